// GNN2_41377714930173
// MI455X (gfx1250) — compile-verified
//
#include <hip/hip_runtime.h>
#include <hip/hip_bf16.h>

// ---------------------------------------------------------------------------
// GATv2Conv + global LayerNorm + mean-pool + linear head, for gfx1250.
// Matrix transforms xl = x@W_l + b_l, xr = x@W_r + b_r use
// V_WMMA_F32_16X16X4_F32 (exact shape match: 16 nodes x K=4 x 16 cols).
// Edge phase is L2-resident (xl/xr/acc = 25.6MB each << 192MB L2).
// ---------------------------------------------------------------------------

typedef __attribute__((ext_vector_type(2))) float v2f;
typedef __attribute__((ext_vector_type(8))) float v8f;

#define HID 64
#define NGRAPH 64

__device__ __forceinline__ unsigned enc_f32(float f) {
    unsigned u = __float_as_uint(f);
    return (u & 0x80000000u) ? ~u : (u | 0x80000000u);
}
__device__ __forceinline__ float dec_f32(unsigned k) {
    unsigned u = (k & 0x80000000u) ? (k & 0x7FFFFFFFu) : ~k;
    return __uint_as_float(u);
}

// ---------------------------------------------------------------------------
// K1: xl/xr via WMMA. One wave handles 16 nodes x full 64 outputs.
// A (16x4 f32, 2 VGPRs): lane l -> M=l%16, K = 2*(l/16) + {0,1}
// B (4x16 f32, 2 VGPRs): lane l -> N=l%16, K = 2*(l/16) + {vgpr}
// C (16x16 f32, 8 VGPRs): lane l, vgpr r -> M = r + 8*(l/16), N = l%16
// All 8 WMMAs issued back-to-back; stores take a wave-uniform fast path
// (unguarded clauses) unless this is the (rare) partial last tile.
// ---------------------------------------------------------------------------
__global__ void k_xform_wmma(const float* __restrict__ x,
                             const float* __restrict__ Wl, const float* __restrict__ bl,
                             const float* __restrict__ Wr, const float* __restrict__ br,
                             float* __restrict__ xl, float* __restrict__ xr,
                             int n_nodes) {
    const int lane   = threadIdx.x & 31;
    const int waveId = (blockIdx.x * blockDim.x + threadIdx.x) >> 5;
    const int nodeBase = waveId * 16;
    if (nodeBase >= n_nodes) return;           // wave-uniform exit

    const int m     = lane & 15;
    const int khalf = (lane >> 4) << 1;        // 0 or 2
    int node = nodeBase + m;
    if (node >= n_nodes) node = n_nodes - 1;   // clamp loads, keep EXEC full

    v2f a;
    a.x = x[node * 4 + khalf + 0];
    a.y = x[node * 4 + khalf + 1];

    v8f cL[4], cR[4];
    float biasL[4], biasR[4];

    #pragma unroll
    for (int t = 0; t < 4; ++t) {
        const int col = t * 16 + m;
        v2f bLm, bRm;
        bLm.x = Wl[(khalf + 0) * HID + col];
        bLm.y = Wl[(khalf + 1) * HID + col];
        bRm.x = Wr[(khalf + 0) * HID + col];
        bRm.y = Wr[(khalf + 1) * HID + col];
        biasL[t] = bl[col];
        biasR[t] = br[col];

        v8f z = {};
        cL[t] = __builtin_amdgcn_wmma_f32_16x16x4_f32(false, a, false, bLm,
                                                      (short)0, z, false, false);
        cR[t] = __builtin_amdgcn_wmma_f32_16x16x4_f32(false, a, false, bRm,
                                                      (short)0, z, false, false);
    }

    const int rowBase = nodeBase + ((lane >> 4) << 3);
    const bool fullTile = (nodeBase + 16 <= n_nodes);   // wave-uniform

    if (fullTile) {
        #pragma unroll
        for (int t = 0; t < 4; ++t) {
            const int col = t * 16 + m;
            #pragma unroll
            for (int r = 0; r < 8; ++r) {
                const size_t idx = (size_t)(rowBase + r) * HID + col;
                xl[idx] = cL[t][r] + biasL[t];
                xr[idx] = cR[t][r] + biasR[t];
            }
        }
    } else {
        #pragma unroll
        for (int t = 0; t < 4; ++t) {
            const int col = t * 16 + m;
            #pragma unroll
            for (int r = 0; r < 8; ++r) {
                const int row = rowBase + r;
                if (row < n_nodes) {
                    const size_t idx = (size_t)row * HID + col;
                    xl[idx] = cL[t][r] + biasL[t];
                    xr[idx] = cR[t][r] + biasR[t];
                }
            }
        }
    }
}

// ---------------------------------------------------------------------------
// K2: per-edge attention logit e = sum_c leakyrelu(xl[src]+xr[dst], .2)*att
//     16 lanes per edge, float4 per lane (coalesced 256B row reads).
//     Segment max via monotonic-uint atomic umax.
// ---------------------------------------------------------------------------
__global__ void k_edge_score(const int* __restrict__ ei,
                             const float* __restrict__ xl, const float* __restrict__ xr,
                             const float* __restrict__ att,
                             float* __restrict__ ebuf, unsigned* __restrict__ menc,
                             int n_edges) {
    const int t = blockIdx.x * blockDim.x + threadIdx.x;
    const int e = t >> 4;
    const int g = t & 15;
    if (e >= n_edges) return;

    const int src = ei[e];
    const int dst = ei[n_edges + e];

    const float4 a4 = *(const float4*)(att + 4 * g);
    const float4 l4 = *(const float4*)(xl + (size_t)src * HID + 4 * g);
    const float4 r4 = *(const float4*)(xr + (size_t)dst * HID + 4 * g);

    float h0 = l4.x + r4.x, h1 = l4.y + r4.y, h2 = l4.z + r4.z, h3 = l4.w + r4.w;
    h0 = (h0 > 0.f) ? h0 : 0.2f * h0;
    h1 = (h1 > 0.f) ? h1 : 0.2f * h1;
    h2 = (h2 > 0.f) ? h2 : 0.2f * h2;
    h3 = (h3 > 0.f) ? h3 : 0.2f * h3;
    float p = h0 * a4.x + h1 * a4.y + h2 * a4.z + h3 * a4.w;

    // butterfly reduce within the 16-lane group
    p += __shfl_xor(p, 1, 16);
    p += __shfl_xor(p, 2, 16);
    p += __shfl_xor(p, 4, 16);
    p += __shfl_xor(p, 8, 16);

    if (g == 0) {
        ebuf[e] = p;
        atomicMax(menc + dst, enc_f32(p));
    }
}

// ---------------------------------------------------------------------------
// K3: ex = exp(e - max[dst]); denom[dst] += ex (atomic f32 add)
// ---------------------------------------------------------------------------
__global__ void k_edge_exp(const int* __restrict__ ei,
                           float* __restrict__ ebuf,
                           const unsigned* __restrict__ menc,
                           float* __restrict__ denom, int n_edges) {
    const int e = blockIdx.x * blockDim.x + threadIdx.x;
    if (e >= n_edges) return;
    const int dst = ei[n_edges + e];
    const float mx = dec_f32(menc[dst]);
    const float ex = __expf(ebuf[e] - mx);
    ebuf[e] = ex;
    atomicAdd(denom + dst, ex);
}

// ---------------------------------------------------------------------------
// K4: acc[dst] += (ex/denom[dst]) * xl[src]   (16 lanes/edge, 4 atomics/lane)
// ---------------------------------------------------------------------------
__global__ void k_edge_agg(const int* __restrict__ ei,
                           const float* __restrict__ xl,
                           const float* __restrict__ ebuf,
                           const float* __restrict__ denom,
                           float* __restrict__ acc, int n_edges) {
    const int t = blockIdx.x * blockDim.x + threadIdx.x;
    const int e = t >> 4;
    const int g = t & 15;
    if (e >= n_edges) return;

    const int src = ei[e];
    const int dst = ei[n_edges + e];
    const float alpha = ebuf[e] / denom[dst];

    const float4 l4 = *(const float4*)(xl + (size_t)src * HID + 4 * g);
    float* o = acc + (size_t)dst * HID + 4 * g;
    atomicAdd(o + 0, alpha * l4.x);
    atomicAdd(o + 1, alpha * l4.y);
    atomicAdd(o + 2, alpha * l4.z);
    atomicAdd(o + 3, alpha * l4.w);
}

// ---------------------------------------------------------------------------
// K5: out = relu(acc + conv_bias); accumulate per-graph pooled sums + counts
//     and global sum / sum-of-squares for the LayerNorm statistics.
//     (LN affine commutes with mean pooling, applied in the epilogue.)
// ---------------------------------------------------------------------------
__global__ void k_node_finalize(const float* __restrict__ acc,
                                const float* __restrict__ conv_bias,
                                const int* __restrict__ batch,
                                float* __restrict__ pooled, float* __restrict__ cnt,
                                float* __restrict__ S, int n_nodes) {
    const int t = blockIdx.x * blockDim.x + threadIdx.x;
    const int node = t >> 4;
    const int g = t & 15;

    float s1 = 0.f, s2 = 0.f;
    if (node < n_nodes) {
        const float4 a = *(const float4*)(acc + (size_t)node * HID + 4 * g);
        const float4 b = *(const float4*)(conv_bias + 4 * g);
        const float o0 = fmaxf(a.x + b.x, 0.f);
        const float o1 = fmaxf(a.y + b.y, 0.f);
        const float o2 = fmaxf(a.z + b.z, 0.f);
        const float o3 = fmaxf(a.w + b.w, 0.f);
        const int bg = batch[node];
        float* p = pooled + bg * HID + 4 * g;
        atomicAdd(p + 0, o0);
        atomicAdd(p + 1, o1);
        atomicAdd(p + 2, o2);
        atomicAdd(p + 3, o3);
        s1 = o0 + o1 + o2 + o3;
        s2 = o0 * o0 + o1 * o1 + o2 * o2 + o3 * o3;
        if (g == 0) atomicAdd(cnt + bg, 1.0f);
    }
    // full-wave (32-lane) reduction, then one atomic per wave
    #pragma unroll
    for (int m = 16; m >= 1; m >>= 1) {
        s1 += __shfl_xor(s1, m, 32);
        s2 += __shfl_xor(s2, m, 32);
    }
    if ((threadIdx.x & 31) == 0) {
        atomicAdd(S + 0, s1);
        atomicAdd(S + 1, s2);
    }
}

// ---------------------------------------------------------------------------
// K6: epilogue — LN stats, per-graph mean, LN affine, linear head, sigmoid.
// ---------------------------------------------------------------------------
__global__ void k_epilogue(const float* __restrict__ pooled,
                           const float* __restrict__ cnt,
                           const float* __restrict__ S,
                           const float* __restrict__ ln_w, const float* __restrict__ ln_b,
                           const float* __restrict__ lin_w, const float* __restrict__ lin_b,
                           float* __restrict__ out, int n_nodes) {
    const int gph = threadIdx.x;
    if (gph >= NGRAPH) return;
    const float M   = (float)n_nodes * (float)HID;
    const float mu  = S[0] / M;
    const float var = S[1] / M - mu * mu;
    const float inv = rsqrtf(var + 1e-5f);
    const float c   = fmaxf(cnt[gph], 1.0f);
    float y = lin_b[0];
    for (int ch = 0; ch < HID; ++ch) {
        const float pv = pooled[gph * HID + ch] / c;
        const float nv = (pv - mu) * inv * ln_w[ch] + ln_b[ch];
        y += nv * lin_w[ch];
    }
    out[gph] = 1.0f / (1.0f + __expf(-y));
}

// ---------------------------------------------------------------------------
static inline size_t alignUp(size_t v, size_t a) { return (v + a - 1) & ~(a - 1); }

extern "C" void kernel_launch(void* const* d_in, const int* in_sizes, int n_in,
                              void* d_out, int out_size, void* d_ws, size_t ws_size,
                              hipStream_t stream) {
    const float* x         = (const float*)d_in[0];
    const int*   ei        = (const int*)  d_in[1];
    const int*   batch     = (const int*)  d_in[2];
    const float* W_l       = (const float*)d_in[3];
    const float* b_l       = (const float*)d_in[4];
    const float* W_r       = (const float*)d_in[5];
    const float* b_r       = (const float*)d_in[6];
    const float* att       = (const float*)d_in[7];
    const float* conv_bias = (const float*)d_in[8];
    const float* ln_w      = (const float*)d_in[9];
    const float* ln_b      = (const float*)d_in[10];
    const float* lin_w     = (const float*)d_in[11];
    const float* lin_b     = (const float*)d_in[12];
    float* out = (float*)d_out;

    const int N = in_sizes[0] / 4;   // nodes
    const int E = in_sizes[1] / 2;   // edges

    // workspace layout
    char* ws = (char*)d_ws;
    size_t off = 0;
    float*    xl    = (float*)(ws + off); off = alignUp(off + (size_t)N * HID * 4, 256);
    float*    xr    = (float*)(ws + off); off = alignUp(off + (size_t)N * HID * 4, 256);
    float*    ebuf  = (float*)(ws + off); off = alignUp(off + (size_t)E * 4, 256);
    const size_t zeroStart = off;
    unsigned* menc  = (unsigned*)(ws + off); off = alignUp(off + (size_t)N * 4, 256);
    float*    denom = (float*)(ws + off); off = alignUp(off + (size_t)N * 4, 256);
    float*    acc   = (float*)(ws + off); off = alignUp(off + (size_t)N * HID * 4, 256);
    float*    pooled= (float*)(ws + off); off = alignUp(off + (size_t)NGRAPH * HID * 4, 256);
    float*    cnt   = (float*)(ws + off); off = alignUp(off + (size_t)NGRAPH * 4, 256);
    float*    S     = (float*)(ws + off); off = alignUp(off + 2 * 4, 256);

    // zero accumulators (menc zero == encoding of the lowest float key)
    hipMemsetAsync(ws + zeroStart, 0, off - zeroStart, stream);

    const int BLK = 256;
    // K1: WMMA transforms (8 waves/block, 16 nodes/wave -> 128 nodes/block)
    {
        int blocks = (N + 127) / 128;
        k_xform_wmma<<<blocks, BLK, 0, stream>>>(x, W_l, b_l, W_r, b_r, xl, xr, N);
    }
    // K2: edge scores + segment max (16 lanes/edge)
    {
        long long th = (long long)E * 16;
        int blocks = (int)((th + BLK - 1) / BLK);
        k_edge_score<<<blocks, BLK, 0, stream>>>(ei, xl, xr, att, ebuf, menc, E);
    }
    // K3: exp + denom
    {
        int blocks = (E + BLK - 1) / BLK;
        k_edge_exp<<<blocks, BLK, 0, stream>>>(ei, ebuf, menc, denom, E);
    }
    // K4: weighted scatter aggregation (16 lanes/edge)
    {
        long long th = (long long)E * 16;
        int blocks = (int)((th + BLK - 1) / BLK);
        k_edge_agg<<<blocks, BLK, 0, stream>>>(ei, xl, ebuf, denom, acc, E);
    }
    // K5: bias+relu, pooled sums, counts, LN stats (16 lanes/node)
    {
        long long th = (long long)N * 16;
        int blocks = (int)((th + BLK - 1) / BLK);
        k_node_finalize<<<blocks, BLK, 0, stream>>>(acc, conv_bias, batch, pooled, cnt, S, N);
    }
    // K6: epilogue
    k_epilogue<<<1, 64, 0, stream>>>(pooled, cnt, S, ln_w, ln_b, lin_w, lin_b, out, N);
}